// SparseMHA_51857435132329
// MI455X (gfx1250) — compile-verified
//
#include <hip/hip_runtime.h>

// Sparse multi-head graph attention (GAT-style) for MI455X / gfx1250.
// N=50000 nodes, E=800000 edges, IN_DIM=128, OUT_DIM=16, H=8 heads.
// fp32 end-to-end; projections use V_WMMA_F32_16X16X4_F32.

#define IN_DIM   128
#define OUT_DIM  16
#define HEADS    8
#define QKV_COLS 128   // OUT_DIM * HEADS

typedef __attribute__((ext_vector_type(2))) float v2f;
typedef __attribute__((ext_vector_type(8))) float v8f;

__device__ __forceinline__ v8f wmma_f32_16x16x4(v2f a, v2f b, v8f c) {
    // 8 args: (neg_a, A, neg_b, B, c_mod, C, reuse_a, reuse_b)
    return __builtin_amdgcn_wmma_f32_16x16x4_f32(false, a, false, b, (short)0, c,
                                                 false, false);
}

// Monotonic uint encoding of float: preserves ordering, enables atomicMax(uint)
// as float segment-max. enc(-inf)=0x007FFFFF; we init buffers to 0 (absolute min).
__device__ __forceinline__ unsigned enc_f32(float f) {
    unsigned u = __float_as_uint(f);
    return (u & 0x80000000u) ? ~u : (u | 0x80000000u);
}
__device__ __forceinline__ float dec_f32(unsigned u) {
    return (u & 0x80000000u) ? __uint_as_float(u & 0x7FFFFFFFu)
                             : __uint_as_float(~u);
}

// ---------------------------------------------------------------- init ------
__global__ void init_kernel(float* __restrict__ z, unsigned* __restrict__ m_enc,
                            float* __restrict__ out, int n) {
    int nz   = n * HEADS;
    int nout = n * QKV_COLS;
    for (int i = blockIdx.x * blockDim.x + threadIdx.x; i < nout;
         i += gridDim.x * blockDim.x) {
        out[i] = 0.0f;
        if (i < nz) { z[i] = 0.0f; m_enc[i] = 0u; }
    }
}

// ------------------------------------------------- QKV projection (WMMA) ----
// grid = (N/16, 3): x = 16-row tile of h, y = which matrix (0=q,1=k,2=v).
// 8 waves per block -> 8 column tiles of 16. K=128 swept in 32 k-steps of 4.
__global__ void __launch_bounds__(256)
qkv_proj_kernel(const float* __restrict__ h,
                const float* __restrict__ Wq, const float* __restrict__ bq,
                const float* __restrict__ Wk, const float* __restrict__ bk,
                const float* __restrict__ Wv, const float* __restrict__ bv,
                float* __restrict__ q, float* __restrict__ k,
                float* __restrict__ v, int n) {
    __shared__ float hs[16 * IN_DIM];

    int r0 = blockIdx.x * 16;
    // Cooperative, coalesced load of the 16x128 h tile into LDS.
    for (int idx = threadIdx.x; idx < 16 * IN_DIM; idx += 256) {
        int rr = idx >> 7, cc = idx & 127;
        int gr = r0 + rr;
        hs[idx] = (gr < n) ? h[(size_t)gr * IN_DIM + cc] : 0.0f;
    }
    __syncthreads();

    const float* W; const float* bias; float* outp; float scale;
    if (blockIdx.y == 0)      { W = Wq; bias = bq; outp = q; scale = 0.25f; } // 16^-0.5
    else if (blockIdx.y == 1) { W = Wk; bias = bk; outp = k; scale = 1.0f; }
    else                      { W = Wv; bias = bv; outp = v; scale = 1.0f; }

    int wave = threadIdx.x >> 5;
    int lane = threadIdx.x & 31;
    int c0   = wave * 16;
    int half = lane >> 4;    // 0: K={0,1}, 1: K={2,3} within each k-step
    int lm   = lane & 15;

    v8f acc = {};
    for (int k0 = 0; k0 < IN_DIM; k0 += 4) {
        int kk = k0 + 2 * half;
        // A fragment 16x4 f32: lanes 0-15 rows, VGPR0/1 = consecutive K.
        v2f a; a.x = hs[lm * IN_DIM + kk];
               a.y = hs[lm * IN_DIM + kk + 1];
        // B fragment 4x16 f32: lanes = columns, VGPR0/1 = consecutive K rows.
        v2f b; b.x = W[(size_t)kk * QKV_COLS + c0 + lm];
               b.y = W[(size_t)(kk + 1) * QKV_COLS + c0 + lm];
        acc = wmma_f32_16x16x4(a, b, acc);
    }

    float bcol = bias[c0 + lm];
#pragma unroll
    for (int j = 0; j < 8; ++j) {
        int r = r0 + j + 8 * half;              // C/D layout: VGPR j -> M=j(+8)
        if (r < n)
            outp[(size_t)r * QKV_COLS + c0 + lm] = (acc[j] + bcol) * scale;
    }
}

// ---------------------------------------------------------------- SDDMM -----
// Wave handles a contiguous edge chunk. Lanes split the two 128-float rows as
// float4s; parity-preserving xor reduction yields the 8 interleaved head dots.
#define SDDMM_EPW 16
__global__ void __launch_bounds__(256)
sddmm_kernel(const float* __restrict__ q, const float* __restrict__ k,
             const int* __restrict__ row, const int* __restrict__ col,
             float* __restrict__ s, unsigned* __restrict__ m_enc, int E) {
    int wave = blockIdx.x * (blockDim.x >> 5) + (threadIdx.x >> 5);
    int lane = threadIdx.x & 31;
    int e0 = wave * SDDMM_EPW;
    int e1 = min(E, e0 + SDDMM_EPW);
    for (int e = e0; e < e1; ++e) {
        int r = row[e], c = col[e];
        if (e + 1 < e1)   // hide random-gather latency of next k row
            __builtin_prefetch(&k[(size_t)col[e + 1] * QKV_COLS + lane * 4], 0, 0);

        float4 q4 = *reinterpret_cast<const float4*>(q + (size_t)r * QKV_COLS + lane * 4);
        float4 k4 = *reinterpret_cast<const float4*>(k + (size_t)c * QKV_COLS + lane * 4);
        float p0 = q4.x * k4.x, p1 = q4.y * k4.y;
        float p2 = q4.z * k4.z, p3 = q4.w * k4.w;
        // Column j = lane*4+i maps to head (4*(lane&1)+i); sum over the 16
        // same-parity lanes with xor strides 2,4,8,16.
#pragma unroll
        for (int off = 2; off < 32; off <<= 1) {
            p0 += __shfl_xor(p0, off, 32);
            p1 += __shfl_xor(p1, off, 32);
            p2 += __shfl_xor(p2, off, 32);
            p3 += __shfl_xor(p3, off, 32);
        }
        if (lane < 2) {   // lane0: heads 0-3, lane1: heads 4-7
            *reinterpret_cast<float4*>(s + (size_t)e * HEADS + 4 * lane) =
                make_float4(p0, p1, p2, p3);
            unsigned* mp = m_enc + (size_t)r * HEADS + 4 * lane;
            atomicMax(mp + 0, enc_f32(p0));
            atomicMax(mp + 1, enc_f32(p1));
            atomicMax(mp + 2, enc_f32(p2));
            atomicMax(mp + 3, enc_f32(p3));
        }
    }
}

// ------------------------------------------------------- softmax numerator --
__global__ void softmax_kernel(float* __restrict__ s, const int* __restrict__ row,
                               const unsigned* __restrict__ m_enc,
                               float* __restrict__ z, int E) {
    long long total = (long long)E * HEADS;
    for (long long idx = (long long)blockIdx.x * blockDim.x + threadIdx.x;
         idx < total; idx += (long long)gridDim.x * blockDim.x) {
        int e = (int)(idx >> 3), hh = (int)(idx & 7);
        int r = row[e];
        float mv = dec_f32(m_enc[(size_t)r * HEADS + hh]);
        float pv = expf(s[idx] - mv);
        s[idx] = pv;
        atomicAdd(&z[(size_t)r * HEADS + hh], pv);
    }
}

// ----------------------------------------------------------------- SPMM -----
// Lanes cover the 128 out columns (float4 each). Register-accumulate while the
// (sorted) row is unchanged; flush with atomicAdd on row change (avg deg ~16
// -> ~16x fewer atomics than per-edge).
#define SPMM_EPW 16
__global__ void __launch_bounds__(256)
spmm_kernel(const float* __restrict__ p, const float* __restrict__ z,
            const float* __restrict__ v, const int* __restrict__ row,
            const int* __restrict__ col, float* __restrict__ out, int E) {
    int wave = blockIdx.x * (blockDim.x >> 5) + (threadIdx.x >> 5);
    int lane = threadIdx.x & 31;
    int e0 = wave * SPMM_EPW;
    int e1 = min(E, e0 + SPMM_EPW);
    if (e0 >= e1) return;

    int par = lane & 1;   // head group of this lane's 4 columns
    float a0 = 0, a1 = 0, a2 = 0, a3 = 0;
    float4 zi = make_float4(0, 0, 0, 0);
    int cur = -1;
    for (int e = e0; e < e1; ++e) {
        int r = row[e], c = col[e];
        if (r != cur) {
            if (cur >= 0) {
                float* op = out + (size_t)cur * QKV_COLS + lane * 4;
                atomicAdd(op + 0, a0); atomicAdd(op + 1, a1);
                atomicAdd(op + 2, a2); atomicAdd(op + 3, a3);
                a0 = a1 = a2 = a3 = 0;
            }
            cur = r;
            float4 zz = *reinterpret_cast<const float4*>(z + (size_t)r * HEADS + 4 * par);
            zi = make_float4(1.0f / zz.x, 1.0f / zz.y, 1.0f / zz.z, 1.0f / zz.w);
        }
        if (e + 1 < e1)
            __builtin_prefetch(&v[(size_t)col[e + 1] * QKV_COLS + lane * 4], 0, 0);

        float4 p4 = *reinterpret_cast<const float4*>(p + (size_t)e * HEADS + 4 * par);
        float4 v4 = *reinterpret_cast<const float4*>(v + (size_t)c * QKV_COLS + lane * 4);
        a0 += p4.x * zi.x * v4.x;
        a1 += p4.y * zi.y * v4.y;
        a2 += p4.z * zi.z * v4.z;
        a3 += p4.w * zi.w * v4.w;
    }
    float* op = out + (size_t)cur * QKV_COLS + lane * 4;
    atomicAdd(op + 0, a0); atomicAdd(op + 1, a1);
    atomicAdd(op + 2, a2); atomicAdd(op + 3, a3);
}

// ---------------------------------------------------------------- launch ----
extern "C" void kernel_launch(void* const* d_in, const int* in_sizes, int n_in,
                              void* d_out, int out_size, void* d_ws, size_t ws_size,
                              hipStream_t stream) {
    const float* h  = (const float*)d_in[0];
    const float* Wq = (const float*)d_in[1];
    const float* bq = (const float*)d_in[2];
    const float* Wk = (const float*)d_in[3];
    const float* bk = (const float*)d_in[4];
    const float* Wv = (const float*)d_in[5];
    const float* bv = (const float*)d_in[6];
    const int*   row = (const int*)d_in[7];
    const int*   col = (const int*)d_in[8];
    float* out = (float*)d_out;

    int n = in_sizes[0] / IN_DIM;   // 50000
    int E = in_sizes[7];            // 800000

    // Workspace carve-up (~105.6 MB; all intermediates fit MI455X's 192MB L2).
    char* ws = (char*)d_ws;
    size_t qkvBytes = (size_t)n * QKV_COLS * sizeof(float);
    float*    q     = (float*)ws;    ws += qkvBytes;
    float*    k     = (float*)ws;    ws += qkvBytes;
    float*    v     = (float*)ws;    ws += qkvBytes;
    float*    s     = (float*)ws;    ws += (size_t)E * HEADS * sizeof(float);
    unsigned* m_enc = (unsigned*)ws; ws += (size_t)n * HEADS * sizeof(unsigned);
    float*    z     = (float*)ws;

    init_kernel<<<512, 256, 0, stream>>>(z, m_enc, out, n);

    dim3 g1((n + 15) / 16, 3);
    qkv_proj_kernel<<<g1, 256, 0, stream>>>(h, Wq, bq, Wk, bk, Wv, bv, q, k, v, n);

    int waves  = (E + SDDMM_EPW - 1) / SDDMM_EPW;
    int blocks = (waves + 7) / 8;
    sddmm_kernel<<<blocks, 256, 0, stream>>>(q, k, row, col, s, m_enc, E);

    softmax_kernel<<<2048, 256, 0, stream>>>(s, row, m_enc, z, E);

    spmm_kernel<<<blocks, 256, 0, stream>>>(s, z, v, row, col, out, E);
}